// GCNLayer_54417235640952
// MI455X (gfx1250) — compile-verified
//
#include <hip/hip_runtime.h>

// GCN layer: out = relu( scatter_add(edge_vals * (X@W)[edge_cols] -> edge_rows) + bias )
// MI455X plan:
//  - X@W on the fp32 WMMA pipe (V_WMMA_F32_16X16X4_F32): exact fp32, matrix unit.
//  - SpMM: support (16MB) + out (16MB) are L2-resident (192MB L2), so random
//    gathers + f32 scatter-atomics run at L2 rates; HBM only streams X/indices once.

typedef float v2f __attribute__((ext_vector_type(2)));
typedef float v8f __attribute__((ext_vector_type(8)));

#define IN_F 64
#define OUT_F 64

// ---------------------------------------------------------------------------
// Kernel 1: support = X @ W using V_WMMA_F32_16X16X4_F32.
// Block = 128 threads = 4 waves. Block b covers rows [b*64, b*64+64);
// wave w covers 16 rows and all 64 output columns (4 accumulators of 16 cols).
//
// fp32 WMMA lane layouts (ISA 7.12.2):
//  A (16x4):  lanes 0-15 hold M=lane, v0=K0,v1=K1; lanes 16-31 hold v0=K2,v1=K3
//  B (4x16):  v0: lanes0-15 = row K0 (N=lane), lanes16-31 = row K1;
//             v1: lanes0-15 = row K2,          lanes16-31 = row K3
//  C/D (16x16 f32): vgpr r, lanes0-15 -> M=r, N=lane; lanes16-31 -> M=8+r
// ---------------------------------------------------------------------------
__global__ void __launch_bounds__(128)
gcn_gemm_wmma(const float* __restrict__ X, const float* __restrict__ W,
              float* __restrict__ S) {
  const int lane = threadIdx.x & 31;
  const int wave = threadIdx.x >> 5;
  const int l15  = lane & 15;
  const int half = lane >> 4;                 // 0: lanes 0-15, 1: lanes 16-31
  const int row0 = blockIdx.x * 64 + wave * 16;

  v8f acc[4];
#pragma unroll
  for (int j = 0; j < 4; ++j) acc[j] = (v8f){0.f,0.f,0.f,0.f,0.f,0.f,0.f,0.f};

  const float* xrow = X + (size_t)(row0 + l15) * IN_F;

#pragma unroll
  for (int k0 = 0; k0 < IN_F; k0 += 4) {
    // A fragment: this lane's row, two consecutive K values (K0/K1 or K2/K3)
    v2f a = *(const v2f*)(xrow + k0 + half * 2);
#pragma unroll
    for (int j = 0; j < 4; ++j) {
      const int n0 = j * 16;
      v2f b;
      b.x = W[(size_t)(k0 + half)     * OUT_F + n0 + l15];  // rows K0/K1
      b.y = W[(size_t)(k0 + 2 + half) * OUT_F + n0 + l15];  // rows K2/K3
      acc[j] = __builtin_amdgcn_wmma_f32_16x16x4_f32(
          /*neg_a=*/false, a, /*neg_b=*/false, b,
          /*c_mod=*/(short)0, acc[j], /*reuse_a=*/false, /*reuse_b=*/false);
    }
  }

#pragma unroll
  for (int j = 0; j < 4; ++j) {
    const int n0 = j * 16;
#pragma unroll
    for (int r = 0; r < 8; ++r) {
      const int m = r + half * 8;
      S[(size_t)(row0 + m) * OUT_F + n0 + l15] = acc[j][r];
    }
  }
}

// ---------------------------------------------------------------------------
// Kernel 2: zero d_out (scatter target must start from 0 every call).
// ---------------------------------------------------------------------------
__global__ void __launch_bounds__(256)
gcn_zero_f4(float4* __restrict__ p, long long n4) {
  long long i = (long long)blockIdx.x * blockDim.x + threadIdx.x;
  if (i < n4) p[i] = make_float4(0.f, 0.f, 0.f, 0.f);
}

// ---------------------------------------------------------------------------
// Kernel 3: COO scatter. 16 threads per edge; each thread handles one float4
// slice of the 64-float feature row -> coalesced 256B gather per edge, then
// 4 global f32 atomic adds into the (L2-resident) output row.
// ---------------------------------------------------------------------------
__global__ void __launch_bounds__(256)
gcn_scatter(const int* __restrict__ erow, const int* __restrict__ ecol,
            const float* __restrict__ eval, const float* __restrict__ S,
            float* __restrict__ out, int nedges) {
  long long idx = (long long)blockIdx.x * blockDim.x + threadIdx.x;
  int e = (int)(idx >> 4);
  int q = (int)(idx & 15);
  if (e >= nedges) return;

  const int   r = erow[e];
  const int   c = ecol[e];
  const float v = eval[e];

  const float4 s = ((const float4*)(S + (size_t)c * OUT_F))[q];
  float* o = out + (size_t)r * OUT_F + q * 4;
  atomicAdd(o + 0, v * s.x);
  atomicAdd(o + 1, v * s.y);
  atomicAdd(o + 2, v * s.z);
  atomicAdd(o + 3, v * s.w);
}

// ---------------------------------------------------------------------------
// Kernel 4: out = relu(out + bias)
// ---------------------------------------------------------------------------
__global__ void __launch_bounds__(256)
gcn_bias_relu(float4* __restrict__ out, const float4* __restrict__ bias4,
              long long n4) {
  long long i = (long long)blockIdx.x * blockDim.x + threadIdx.x;
  if (i >= n4) return;
  float4 o = out[i];
  float4 b = bias4[i & 15];   // OUT_F/4 == 16 float4 per row
  o.x = fmaxf(o.x + b.x, 0.f);
  o.y = fmaxf(o.y + b.y, 0.f);
  o.z = fmaxf(o.z + b.z, 0.f);
  o.w = fmaxf(o.w + b.w, 0.f);
  out[i] = o;
}

extern "C" void kernel_launch(void* const* d_in, const int* in_sizes, int n_in,
                              void* d_out, int out_size, void* d_ws, size_t ws_size,
                              hipStream_t stream) {
  const float* X    = (const float*)d_in[0];
  const int*   erow = (const int*)  d_in[1];
  const int*   ecol = (const int*)  d_in[2];
  const float* ev   = (const float*)d_in[3];
  const float* W    = (const float*)d_in[4];
  const float* bias = (const float*)d_in[5];
  float* out = (float*)d_out;
  float* S   = (float*)d_ws;          // support = X@W, N*OUT_F floats (16 MB)

  const int N = in_sizes[0] / IN_F;   // 65536
  const int E = in_sizes[1];          // 1048576

  // 1) support = X @ W (fp32 WMMA)
  gcn_gemm_wmma<<<N / 64, 128, 0, stream>>>(X, W, S);

  // 2) out = 0
  const long long n4 = (long long)N * OUT_F / 4;
  gcn_zero_f4<<<(unsigned)((n4 + 255) / 256), 256, 0, stream>>>((float4*)out, n4);

  // 3) scatter-add edges
  const long long nt = (long long)E * 16;
  gcn_scatter<<<(unsigned)((nt + 255) / 256), 256, 0, stream>>>(erow, ecol, ev, S, out, E);

  // 4) bias + relu
  gcn_bias_relu<<<(unsigned)((n4 + 255) / 256), 256, 0, stream>>>(
      (float4*)out, (const float4*)bias, n4);
}